// LoralizedEmbedding_17540646436900
// MI455X (gfx1250) — compile-verified
//
#include <hip/hip_runtime.h>

typedef __attribute__((ext_vector_type(2))) float v2f;
typedef __attribute__((ext_vector_type(8))) float v8f;

#define DIM  1024
#define RANK 16

// One block = one 16-token tile. 8 waves/block; each wave covers 128 dims
// (8 chunks of 16). Per chunk: gather C (orig_weight rows), 4 chained
// v_wmma_f32_16x16x4_f32 add the rank-16 LoRA correction, store D.
__global__ __launch_bounds__(256) void lora_embed_kernel(
    const int*   __restrict__ x,
    const float* __restrict__ W,    // orig_weight [VOCAB, DIM]
    const float* __restrict__ A1,   // aw1 [VOCAB, RANK]
    const float* __restrict__ A2,   // aw2 [RANK, DIM]
    float*       __restrict__ out)  // [NTOK, DIM]
{
    const int lane = threadIdx.x & 31;
    const int wave = threadIdx.x >> 5;
    const int half = lane >> 4;        // 0: lanes 0-15, 1: lanes 16-31
    const int n    = lane & 15;        // column within 16-dim chunk
    const int tokBase = blockIdx.x * 16;

    // ---- A fragments (loop-invariant): 16x4 f32 layout ----
    // lane m (<16): V0=K0,V1=K1 ; lane m+16: V0=K2,V1=K3   (per K-block j)
    const int   tokm = x[tokBase + n];
    const float* arow = A1 + (size_t)tokm * RANK;
    v2f afrag[4];
#pragma unroll
    for (int j = 0; j < 4; ++j) {
        const int k0 = 4 * j + 2 * half;
        afrag[j].x = arow[k0];
        afrag[j].y = arow[k0 + 1];
    }

    // ---- Row pointers for C/D: VGPR r <-> token M = r + 8*half ----
    const float* wrow[8];
    size_t obase[8];
#pragma unroll
    for (int r = 0; r < 8; ++r) {
        const int t = tokBase + half * 8 + r;
        wrow[r]  = W + (size_t)x[t] * DIM;
        obase[r] = (size_t)t * DIM;
    }

    const int dim0 = wave * 128;
#pragma unroll
    for (int c = 0; c < 8; ++c) {
        const int dim = dim0 + c * 16 + n;

        // C = gathered orig_weight values (each half-wave: 64B contiguous/row)
        v8f acc;
#pragma unroll
        for (int r = 0; r < 8; ++r) acc[r] = wrow[r][dim];

        // 4 chained K=4 WMMAs cover RANK=16 in exact fp32.
        // B 4x16 layout: V0 = rows (4j+0 | 4j+2), V1 = rows (4j+1 | 4j+3)
#pragma unroll
        for (int j = 0; j < 4; ++j) {
            const int k0 = 4 * j + 2 * half;
            v2f b;
            b.x = A2[(size_t)k0 * DIM + dim];
            b.y = A2[(size_t)(k0 + 1) * DIM + dim];
            acc = __builtin_amdgcn_wmma_f32_16x16x4_f32(
                /*neg_a=*/false, afrag[j], /*neg_b=*/false, b,
                /*c_mod=*/(short)0, acc, /*reuse_a=*/false, /*reuse_b=*/false);
        }

        // Streamed output: nontemporal to preserve L2 for weight-row reuse.
#pragma unroll
        for (int r = 0; r < 8; ++r)
            __builtin_nontemporal_store(acc[r], out + obase[r] + dim);
    }
}

extern "C" void kernel_launch(void* const* d_in, const int* in_sizes, int n_in,
                              void* d_out, int out_size, void* d_ws, size_t ws_size,
                              hipStream_t stream) {
    const int*   x  = (const int*)d_in[0];    // [4,4096] int32
    const float* W  = (const float*)d_in[1];  // [128000,1024] f32
    const float* A1 = (const float*)d_in[2];  // [128000,16]  f32
    const float* A2 = (const float*)d_in[3];  // [16,1024]    f32
    float* out = (float*)d_out;               // [4,4096,1024] f32

    const int ntok  = in_sizes[0];   // 16384
    const int tiles = ntok / 16;     // 1024 tiles of 16 tokens
    lora_embed_kernel<<<tiles, 256, 0, stream>>>(x, W, A1, A2, out);
}